// QRegulariser_65481071406581
// MI455X (gfx1250) — compile-verified
//
#include <hip/hip_runtime.h>

typedef float v2f __attribute__((ext_vector_type(2)));
typedef float v4f __attribute__((ext_vector_type(4)));
typedef float v8f __attribute__((ext_vector_type(8)));
typedef int   v4i __attribute__((ext_vector_type(4)));

typedef __attribute__((address_space(1))) v4i gv4i;   // global int4
typedef __attribute__((address_space(3))) v4i sv4i;   // LDS int4

#define NROWS 65536
#define H     1024
#define Q     8
#define ROWS_PER_WAVE  16
#define WAVES          8
#define ROWS_PER_BLOCK (ROWS_PER_WAVE * WAVES)   // 128
#define NBLOCKS        (NROWS / ROWS_PER_BLOCK)  // 512
#define KCHUNK         64
#define KSTEPS         (KCHUNK / 4)              // 16
#define NCHUNKS        (H / KCHUNK)              // 16
#define APAD           68    // padded LDS row stride (floats); 68 % 64 == 4 -> conflict-free b64 reads
#define BUF_FLOATS     (ROWS_PER_WAVE * APAD)    // 1088 floats per buffer

// LDS layout (floats):
//   [0, 8192)                        W  (8 x 1024)
//   [8192, 8192 + 8*2*BUF_FLOATS)    per-wave double-buffered A tiles
//   [..., +8)                        per-wave partial sums
#define LDS_FLOATS (Q*H + WAVES*2*BUF_FLOATS + WAVES)

// Issue 8 async global->LDS copies staging a 16 x KCHUNK tile (2 rows per op).
__device__ __forceinline__ void stage_async(const float* __restrict__ hs,
                                            int rowBase, int kc,
                                            float* __restrict__ buf, int lane)
{
    const int rsub = lane >> 4;         // 0 or 1: which of the two rows
    const int cl   = (lane & 15) * 4;   // column within chunk (floats)
    #pragma unroll
    for (int r = 0; r < ROWS_PER_WAVE; r += 2) {
        const float* g = hs + (size_t)(rowBase + r + rsub) * H + kc + cl;
        float*       l = buf + (r + rsub) * APAD + cl;
        __builtin_amdgcn_global_load_async_to_lds_b128(
            (gv4i*)(void*)g, (sv4i*)(void*)l, /*offset=*/0, /*cpol=*/0);
    }
}

__global__ __launch_bounds__(256)
void qreg_gemm_kernel(const float* __restrict__ hs,
                      const float* __restrict__ W,
                      const float* __restrict__ bias,
                      float* __restrict__ partial)
{
    extern __shared__ float lds[];
    float* ldsW = lds;                                   // 8192 floats
    float* ldsA = lds + Q * H;                           // double buffers
    float* ldsR = ldsA + WAVES * 2 * BUF_FLOATS;         // 8 floats

    const int tid  = threadIdx.x;
    const int wave = tid >> 5;
    const int lane = tid & 31;
    const int m    = lane & 15;       // A row / C column index
    const int kk   = (lane >> 4) * 2; // K sub-offset for this half-wave
    const int nm   = lane & 7;        // qubit index (W rows replicated for n>=8)

    // ---- stage all of W [8 x 1024] into LDS, coalesced float4 ----
    {
        const v4f* Wg = (const v4f*)W;
        v4f*       Wl = (v4f*)ldsW;
        #pragma unroll
        for (int i = 0; i < (Q * H / 4) / 256; ++i)
            Wl[tid + i * 256] = Wg[tid + i * 256];
    }
    __syncthreads();

    const int rowBase = blockIdx.x * ROWS_PER_BLOCK + wave * ROWS_PER_WAVE;
    float* buf0 = ldsA + wave * 2 * BUF_FLOATS;
    float* buf1 = buf0 + BUF_FLOATS;

    // prologue: prefetch chunk 0
    stage_async(hs, rowBase, 0, buf0, lane);

    v8f acc = {};   // C/D 16x16 f32 accumulator (8 VGPRs)
    float* cur = buf0;
    float* nxt = buf1;

    for (int c = 0; c < NCHUNKS; ++c) {
        if (c + 1 < NCHUNKS) {
            // prefetch next chunk into the other buffer, then wait only for
            // the OLDER 8 copies (in-order completion) -> current tile ready
            stage_async(hs, rowBase, (c + 1) * KCHUNK, nxt, lane);
            __builtin_amdgcn_s_wait_asynccnt(8);
        } else {
            __builtin_amdgcn_s_wait_asynccnt(0);
        }

        const float* bW = ldsW + nm * H + c * KCHUNK;
        #pragma unroll
        for (int ks = 0; ks < KSTEPS; ++ks) {
            // A-fragment 16x4 (ds_load_b64, conflict-free via APAD)
            v2f a = *(const v2f*)(cur + m * APAD + ks * 4 + kk);
            // B-fragment 4x16 (8-lane LDS broadcast)
            v2f b = *(const v2f*)(bW + ks * 4 + kk);
            acc = __builtin_amdgcn_wmma_f32_16x16x4_f32(
                      /*neg_a=*/false, a, /*neg_b=*/false, b,
                      /*c_mod=*/(short)0, acc, /*reuse_a=*/false, /*reuse_b=*/false);
        }
        float* t = cur; cur = nxt; nxt = t;
    }

    // ---- epilogue: theta -> cos^2 -> product over 8 qubits -> sum(1 - p0) ----
    // C layout: lanes 0-15 hold rows j (VGPR j), lanes 16-31 hold rows j+8.
    // Columns 8-15 duplicate 0-7, so a 3-level XOR butterfly over the low
    // 3 lane bits yields the 8-qubit product on every lane.
    const float bn = bias[nm];
    float s = 0.0f;
    #pragma unroll
    for (int j = 0; j < 8; ++j) {
        float th = acc[j] + bn;
        float c  = __cosf(0.5f * th);
        float p  = c * c;
        p *= __shfl_xor(p, 1, 32);
        p *= __shfl_xor(p, 2, 32);
        p *= __shfl_xor(p, 4, 32);
        s += 1.0f - p;        // lanes 0-15: rows 0-7 ; lanes 16-31: rows 8-15
    }
    s += __shfl_xor(s, 16, 32);   // full 16-row tile sum on every lane

    if (lane == 0) ldsR[wave] = s;
    __syncthreads();
    if (tid == 0) {
        float t = 0.0f;
        #pragma unroll
        for (int w = 0; w < WAVES; ++w) t += ldsR[w];
        partial[blockIdx.x] = t;   // deterministic per-block partial
    }
}

__global__ __launch_bounds__(256)
void qreg_reduce_kernel(const float* __restrict__ partial, float* __restrict__ out)
{
    __shared__ float r[8];
    const int tid = threadIdx.x;
    float v = partial[tid] + partial[tid + 256];   // NBLOCKS == 512
    v += __shfl_xor(v, 1, 32);
    v += __shfl_xor(v, 2, 32);
    v += __shfl_xor(v, 4, 32);
    v += __shfl_xor(v, 8, 32);
    v += __shfl_xor(v, 16, 32);
    if ((tid & 31) == 0) r[tid >> 5] = v;
    __syncthreads();
    if (tid < 8) {
        float t = r[tid];
        t += __shfl_xor(t, 1, 32);
        t += __shfl_xor(t, 2, 32);
        t += __shfl_xor(t, 4, 32);
        if (tid == 0) out[0] = t * (1.0f / (float)NROWS);
    }
}

extern "C" void kernel_launch(void* const* d_in, const int* in_sizes, int n_in,
                              void* d_out, int out_size, void* d_ws, size_t ws_size,
                              hipStream_t stream)
{
    const float* hs   = (const float*)d_in[0];  // [65536, 1024] f32
    const float* W    = (const float*)d_in[1];  // [8, 1024] f32
    const float* bias = (const float*)d_in[2];  // [8] f32
    float* partial = (float*)d_ws;              // 512 floats of scratch

    const size_t ldsBytes = (size_t)LDS_FLOATS * sizeof(float);  // ~102 KB (<< 320 KB WGP LDS)
    qreg_gemm_kernel<<<NBLOCKS, 256, ldsBytes, stream>>>(hs, W, bias, partial);
    qreg_reduce_kernel<<<1, 256, 0, stream>>>(partial, (float*)d_out);
}